// gdfn_region_batch_73194832658614
// MI455X (gfx1250) — compile-verified
//
#include <hip/hip_runtime.h>
#include <stdint.h>

// ---- problem dims (fixed by reference setup_inputs) ----
#define NB 8
#define NC 3
#define NH 512
#define NW 512
#define HP (NH + 2)   // padded (noise) height
#define WP (NW + 2)   // padded (noise) width

// ---- tiling: 64x4 pixels per 256-thread (8 x wave32) block ----
#define TX 64
#define TY 4
#define TPX (TX + 2)      // 66
#define TPY (TY + 2)      // 6
#define RTSZ (TPY * TPX)  // 396 (one padded plane)
#define PTSZ (NC * RTSZ)  // 1188 (3 channels)
#define NTHREADS 256

#if defined(__AMDGCN__) && __has_builtin(__builtin_amdgcn_tensor_load_to_lds)
#define HAVE_TDM 1
#else
#define HAVE_TDM 0
#endif

#if HAVE_TDM
typedef unsigned int tdm_v4u __attribute__((ext_vector_type(4)));
typedef int tdm_v8i __attribute__((ext_vector_type(8)));
typedef int tdm_v4i __attribute__((ext_vector_type(4)));

// One TDM DMA: 3D tile (x=TPX, y=TPY, z=NC) of the padded-noise tensor
// (plane [HP][WP], channel stride HP*WP) starting at (b, c=0, y0, x0),
// packed contiguously into LDS at `lds`. All addresses in-bounds by
// construction (noise is pre-padded), so no OOB clipping is relied upon.
__device__ __forceinline__ void tdm_noise_tile(const float* gsrc, float* lds) {
  uint64_t ga = (uint64_t)(uintptr_t)gsrc;
  unsigned la = (unsigned)(uintptr_t)lds;  // low 32 bits of flat LDS ptr == LDS byte offset
  const uint64_t st1 = (uint64_t)HP * (uint64_t)WP;  // channel stride (elements)
  tdm_v4u g0 = {
      1u,                                                  // count=1 (valid user D#)
      la,                                                  // [63:32] lds_addr
      (unsigned)ga,                                        // global_addr lo32
      ((unsigned)(ga >> 32) & 0x01FFFFFFu) | 0x80000000u   // addr[56:32] | type=2<<30
  };
  tdm_v8i g1 = {
      (int)(2u << 16),                                     // data_size=2 -> 4 bytes
      (int)(((unsigned)WP & 0xFFFFu) << 16),               // tensor_dim0[15:0]
      (int)((((unsigned)WP >> 16) & 0xFFFFu) |
            (((unsigned)HP & 0xFFFFu) << 16)),             // dim0 hi | dim1 lo
      (int)((((unsigned)HP >> 16) & 0xFFFFu) |
            ((unsigned)TPX << 16)),                        // dim1 hi | tile_dim0
      (int)((unsigned)TPY | ((unsigned)NC << 16)),         // tile_dim1 | tile_dim2
      (int)(unsigned)WP,                                   // dim0_stride lo32
      (int)(((unsigned)(st1 & 0xFFFFu)) << 16),            // dim0_stride hi16(=0) | dim1_stride[15:0]
      (int)(unsigned)(st1 >> 16)                           // dim1_stride[47:16]
  };
  tdm_v4i g2 = { NC, 0, 0, 0 };                            // tensor_dim2 = NC
  tdm_v4i g3 = { 0, 0, 0, 0 };
  tdm_v8i gx = { 0, 0, 0, 0, 0, 0, 0, 0 };                 // extra group (clang-23 6-arg form)
  __builtin_amdgcn_tensor_load_to_lds(g0, g1, g2, g3, gx, 0);
}
#endif

// Issue the padded-noise tile DMA (wave 0 only; TDM issues per-wave, EXEC ignored).
__device__ __forceinline__ void stage_p_issue(float* sp, const float* noise,
                                              int b, int y0, int x0, int tid) {
#if HAVE_TDM
  if (tid < 32) {
    tdm_noise_tile(noise + ((size_t)b * NC * HP + (size_t)y0) * WP + x0, sp);
    __builtin_amdgcn_s_wait_tensorcnt(0);
  }
#else
  (void)sp; (void)noise; (void)b; (void)y0; (void)x0; (void)tid;
#endif
}

// Finish img_p tile: add bounds-checked img into the staged noise
// (zero-padding of img is implicit: out-of-image cells keep pure noise).
__device__ __forceinline__ void stage_p_finish(float* sp, const float* img, const float* noise,
                                               int b, int y0, int x0, int tid) {
  __syncthreads();  // TDM data + any sibling LDS staging visible to all waves
  for (int idx = tid; idx < PTSZ; idx += NTHREADS) {
    int c = idx / RTSZ;
    int rem = idx - c * RTSZ;
    int r = rem / TPX;
    int col = rem - r * TPX;
    int yp = y0 + r, xp = x0 + col;  // padded coords
    bool inb = (yp >= 1) && (yp <= NH) && (xp >= 1) && (xp <= NW);
#if HAVE_TDM
    (void)noise;
    if (inb)
      sp[idx] += img[(((size_t)b * NC + c) * NH + (yp - 1)) * NW + (xp - 1)];
#else
    float v = noise[(((size_t)b * NC + c) * HP + yp) * WP + xp];
    if (inb) v += img[(((size_t)b * NC + c) * NH + (yp - 1)) * NW + (xp - 1)];
    sp[idx] = v;
#endif
  }
  __syncthreads();
}

// ---------------- Kernel 1: img_r ----------------
__global__ __launch_bounds__(NTHREADS) void k1_imgr(
    const float* __restrict__ img, const float* __restrict__ noise,
    const int* __restrict__ sel, float* __restrict__ img_r) {
  const int b = blockIdx.z;
  if (sel[b] == 0) return;  // uniform per block
  __shared__ __align__(16) float sp[PTSZ];
  const int x0 = blockIdx.x * TX, y0 = blockIdx.y * TY;
  const int tid = threadIdx.x;
  stage_p_issue(sp, noise, b, y0, x0, tid);
  stage_p_finish(sp, img, noise, b, y0, x0, tid);

  const int tx = tid & (TX - 1), ty = tid >> 6;
  const int base = ty * TPX + tx;
  float v[NC][9], wmin[9];
#pragma unroll
  for (int k = 0; k < 9; ++k) wmin[k] = 3.4e38f;
#pragma unroll
  for (int c = 0; c < NC; ++c) {
    const float* p = sp + c * RTSZ + base;
#pragma unroll
    for (int i = 0; i < 3; ++i)
#pragma unroll
      for (int j = 0; j < 3; ++j) v[c][i * 3 + j] = p[i * TPX + j];
    float s = 0.f;
#pragma unroll
    for (int k = 0; k < 9; ++k) s += v[c][k];
    const float mean = s * (1.f / 9.f);
    float q = 0.f;
#pragma unroll
    for (int k = 0; k < 9; ++k) { float d = v[c][k] - mean; q += d * d; }
    const float var = q * (1.f / 8.f);  // ddof=1, == std^2
    const float inv = 0.5f / var;
#pragma unroll
    for (int k = 0; k < 9; ++k) {
      float d = v[c][k] - mean;
      wmin[k] = fminf(wmin[k], expf(-(d * d) * inv));
    }
  }
  float num = 0.f, den = 0.f;
#pragma unroll
  for (int k = 0; k < 9; ++k) {
    den += wmin[k];
    num += (v[0][k] + v[1][k] + v[2][k]) * wmin[k];
  }
  img_r[((size_t)b * NH + (y0 + ty)) * NW + (x0 + tx)] = num / den;
}

// ---------------- Kernel 2: img_3d_mod ----------------
__global__ __launch_bounds__(NTHREADS) void k2_mod(
    const float* __restrict__ img, const float* __restrict__ noise,
    const int* __restrict__ sel, const int* __restrict__ mixsel,
    const float* __restrict__ img_r, float* __restrict__ i3m) {
  const int b = blockIdx.z;
  if (sel[b] == 0) return;
  __shared__ __align__(16) float sp[PTSZ];
  __shared__ float rt[RTSZ];
  const int x0 = blockIdx.x * TX, y0 = blockIdx.y * TY;
  const int tid = threadIdx.x;
  stage_p_issue(sp, noise, b, y0, x0, tid);
  // img_r halo tile (zero pad, matching jnp.pad: zeros DO join argmin/argmax)
  for (int idx = tid; idx < RTSZ; idx += NTHREADS) {
    int r = idx / TPX, col = idx - r * TPX;
    int gy = y0 + r - 1, gx = x0 + col - 1;
    float v = 0.f;
    if (gy >= 0 && gy < NH && gx >= 0 && gx < NW)
      v = img_r[((size_t)b * NH + gy) * NW + gx];
    rt[idx] = v;
  }
  stage_p_finish(sp, img, noise, b, y0, x0, tid);

  const int tx = tid & (TX - 1), ty = tid >> 6;
  const int base = ty * TPX + tx;
  float m[9];
#pragma unroll
  for (int i = 0; i < 3; ++i)
#pragma unroll
    for (int j = 0; j < 3; ++j) m[i * 3 + j] = rt[base + i * TPX + j];
  int idmin = 0, idmax = 0;
  float bmin = m[0], bmax = m[0];
#pragma unroll
  for (int k = 1; k < 9; ++k) {              // strict compares -> first occurrence
    if (m[k] < bmin) { bmin = m[k]; idmin = k; }
    if (m[k] > bmax) { bmax = m[k]; idmax = k; }
  }
  const int id = (mixsel[b] != 0) ? idmax : idmin;
  float g = 3.4e38f;
#pragma unroll
  for (int c = 0; c < NC; ++c) {
    const float* p = sp + c * RTSZ + base;
    float v[9];
#pragma unroll
    for (int i = 0; i < 3; ++i)
#pragma unroll
      for (int j = 0; j < 3; ++j) v[i * 3 + j] = p[i * TPX + j];
    float s = 0.f;
#pragma unroll
    for (int k = 0; k < 9; ++k) s += v[k];
    const float mean = s * (1.f / 9.f);
    float q = 0.f;
#pragma unroll
    for (int k = 0; k < 9; ++k) { float d = v[k] - mean; q += d * d; }
    const float var = q * (1.f / 8.f);
    float vd = v[0];
#pragma unroll
    for (int k = 1; k < 9; ++k) vd = (id == k) ? v[k] : vd;  // select v[id]
    float d = vd - mean;
    g = fminf(g, expf(-(d * d) * (0.5f / var)));
  }
  i3m[((size_t)b * NH + (y0 + ty)) * NW + (x0 + tx)] = g;
}

// ---------------- Kernel 3: final blend ----------------
__global__ __launch_bounds__(NTHREADS) void k3_out(
    const float* __restrict__ img, const float* __restrict__ noise,
    const float* __restrict__ remask, const int* __restrict__ sel,
    const float* __restrict__ i3m, float* __restrict__ out) {
  const int b = blockIdx.z;
  const int x0 = blockIdx.x * TX, y0 = blockIdx.y * TY;
  const int tid = threadIdx.x;
  const int tx = tid & (TX - 1), ty = tid >> 6;
  const int y = y0 + ty, x = x0 + tx;
  if (sel[b] == 0) {  // pass-through batches: just copy img
#pragma unroll
    for (int c = 0; c < NC; ++c) {
      size_t gi = (((size_t)b * NC + c) * NH + y) * NW + x;
      out[gi] = img[gi];
    }
    return;
  }
  __shared__ __align__(16) float sp[PTSZ];
  __shared__ float mt[RTSZ];
  stage_p_issue(sp, noise, b, y0, x0, tid);
  // img_3d_mod halo tile (zero pad)
  for (int idx = tid; idx < RTSZ; idx += NTHREADS) {
    int r = idx / TPX, col = idx - r * TPX;
    int gy = y0 + r - 1, gx = x0 + col - 1;
    float v = 0.f;
    if (gy >= 0 && gy < NH && gx >= 0 && gx < NW)
      v = i3m[((size_t)b * NH + gy) * NW + gx];
    mt[idx] = v;
  }
  stage_p_finish(sp, img, noise, b, y0, x0, tid);

  const int base = ty * TPX + tx;
  float m[9], s = 0.f;
#pragma unroll
  for (int i = 0; i < 3; ++i)
#pragma unroll
    for (int j = 0; j < 3; ++j) {
      float mv = mt[base + i * TPX + j];
      m[i * 3 + j] = mv;
      s += mv;
    }
#pragma unroll
  for (int c = 0; c < NC; ++c) {
    const float* p = sp + c * RTSZ + base;
    float t = 0.f;
#pragma unroll
    for (int i = 0; i < 3; ++i)
#pragma unroll
      for (int j = 0; j < 3; ++j) t += p[i * TPX + j] * m[i * 3 + j];
    const float imod = t / s;
    const size_t gi = (((size_t)b * NC + c) * NH + y) * NW + x;
    const float rm = remask[gi];
    out[gi] = imod * (1.f - rm) + img[gi] * rm;
  }
}

extern "C" void kernel_launch(void* const* d_in, const int* in_sizes, int n_in,
                              void* d_out, int out_size, void* d_ws, size_t ws_size,
                              hipStream_t stream) {
  (void)in_sizes; (void)n_in; (void)out_size; (void)ws_size;
  const float* img    = (const float*)d_in[0];  // [8,3,512,512] f32
  const float* noise  = (const float*)d_in[1];  // [8,3,514,514] f32 (pre-padded)
  const float* remask = (const float*)d_in[2];  // [8,3,512,512] f32
  const int*   sel    = (const int*)d_in[3];    // [8] bool->int
  const int*   mixsel = (const int*)d_in[4];    // [8] bool->int
  float* out   = (float*)d_out;
  float* img_r = (float*)d_ws;                        // NB*NH*NW f32 (8.4 MB)
  float* i3m   = img_r + (size_t)NB * NH * NW;        // NB*NH*NW f32 (8.4 MB)

  dim3 grid(NW / TX, NH / TY, NB);  // (8, 128, 8)
  dim3 blk(NTHREADS);
  hipLaunchKernelGGL(k1_imgr, grid, blk, 0, stream, img, noise, sel, img_r);
  hipLaunchKernelGGL(k2_mod,  grid, blk, 0, stream, img, noise, sel, mixsel, img_r, i3m);
  hipLaunchKernelGGL(k3_out,  grid, blk, 0, stream, img, noise, remask, sel, i3m, out);
}